// LeBRS_14147622273112
// MI455X (gfx1250) — compile-verified
//
#include <hip/hip_runtime.h>

// CDNA5 (gfx1250) wave32 WMMA implementation of a tiny fp32 MLP (6->8->8->6)
// over 8.4M rows. Memory-bound (AI ~7 FLOP/B); math done with
// V_WMMA_F32_16X16X4_F32 on transposed tiles so activations never need a
// full transpose -- only a SWAPX16 ds_swizzle between layers.

typedef __attribute__((ext_vector_type(2))) float v2f;
typedef __attribute__((ext_vector_type(8))) float v8f;

static __device__ __forceinline__ float exilu_f(float t) {
    // faithful form: keep if (t >= 1001) || (t <= 447113); NaN -> 0
    return ((t >= 1001.0f) || (t <= 447113.0f)) ? t : 0.0f;
}

// lanes 16-31 fetch the value held by lane-16 (SWAPX16): xor=0x10, or=0, and=0x1f
static __device__ __forceinline__ float swap16(float v) {
    int i = __builtin_bit_cast(int, v);
    i = __builtin_amdgcn_ds_swizzle(i, 0x401F);
    return __builtin_bit_cast(float, i);
}

// D-layout -> B-layout lane mix: lanes 0-15 take 'a' in place,
// lanes 16-31 take lane-16's value of 'bhi'.
static __device__ __forceinline__ float bsel(bool lo, float a, float bhi) {
    float s = swap16(bhi);
    return lo ? a : s;
}

__global__ __launch_bounds__(256) void LeBRS_mlp_wmma(
    const float* __restrict__ x,
    const float* __restrict__ W1, const float* __restrict__ B1,
    const float* __restrict__ W2, const float* __restrict__ B2,
    const float* __restrict__ W3, const float* __restrict__ B3,
    float* __restrict__ out, int num_tiles, long rows)
{
    const int  lane = (int)(threadIdx.x & 31u);
    const int  row  = lane & 15;
    const bool lo   = lane < 16;

    // ---- Weights as WMMA A-matrices (f32 16x4 layout), loaded once. ----
    // A layout: lanes 0-15 hold (K=0,K=1) of row M=lane; lanes 16-31 hold (K=2,K=3).
    v2f a1c0 = {0.f, 0.f}, a1c1 = {0.f, 0.f};   // W1: 8x6, K chunks [0..3],[4..7]
    v2f a2c0 = {0.f, 0.f}, a2c1 = {0.f, 0.f};   // W2: 8x8
    v2f a3c0 = {0.f, 0.f}, a3c1 = {0.f, 0.f};   // W3: 6x8
    if (row < 8) {
        const float* w = W1 + row * 6;
        if (lo) { a1c0.x = w[0]; a1c0.y = w[1]; a1c1.x = w[4]; a1c1.y = w[5]; }
        else    { a1c0.x = w[2]; a1c0.y = w[3]; /* K=6,7 padded zero */ }
        w = W2 + row * 8;
        if (lo) { a2c0.x = w[0]; a2c0.y = w[1]; a2c1.x = w[4]; a2c1.y = w[5]; }
        else    { a2c0.x = w[2]; a2c0.y = w[3]; a2c1.x = w[6]; a2c1.y = w[7]; }
    }
    if (row < 6) {
        const float* w = W3 + row * 8;
        if (lo) { a3c0.x = w[0]; a3c0.y = w[1]; a3c1.x = w[4]; a3c1.y = w[5]; }
        else    { a3c0.x = w[2]; a3c0.y = w[3]; a3c1.x = w[6]; a3c1.y = w[7]; }
    }

    // ---- Biases as C accumulators (D layout: VGPR r -> M=r lanes 0-15). ----
    v8f c1 = {}, c2 = {}, c3 = {};
    if (lo) {
#pragma unroll
        for (int r = 0; r < 8; ++r) { c1[r] = B1[r]; c2[r] = B2[r]; }
#pragma unroll
        for (int r = 0; r < 6; ++r) { c3[r] = B3[r]; }
    }

    const int wave   = (int)((blockIdx.x * blockDim.x + threadIdx.x) >> 5);
    const int nwaves = (int)((gridDim.x * blockDim.x) >> 5);

    for (int tile = wave; tile < num_tiles; tile += nwaves) {
        const long r0 = (long)tile * 16 + row;
        const long rc = (r0 < rows) ? r0 : (rows - 1);
        const float* xr = x + rc * 6;

        // Prefetch next grid-stride tile's cacheline (global_prefetch_b8).
        {
            const long rn = (long)(tile + nwaves) * 16 + row;
            if (rn < rows) __builtin_prefetch(x + rn * 6, 0, 0);
        }

        // B = x^T (K=6 padded). B layout: lanes 0-15 -> K=v, lanes 16-31 -> K=v+2.
        // chunk0: lo lanes need x[r][0],x[r][1]; hi lanes x[r][2],x[r][3].
        v2f bb0 = __builtin_nontemporal_load((const v2f*)(xr + (lo ? 0 : 2)));
        // chunk1: lo lanes x[r][4],x[r][5]; hi lanes K=6,7 -> zero pad.
        v2f t45 = __builtin_nontemporal_load((const v2f*)(xr + 4));
        v2f bb1; bb1.x = lo ? t45.x : 0.f; bb1.y = lo ? t45.y : 0.f;

        // ---- Layer 1: D1 = W1 * x^T + b1  (two K=4 chunks) ----
        v8f d = __builtin_amdgcn_wmma_f32_16x16x4_f32(
            false, a1c0, false, bb0, (short)0, c1, false, false);
        d = __builtin_amdgcn_wmma_f32_16x16x4_f32(
            false, a1c1, false, bb1, (short)0, d, false, false);
#pragma unroll
        for (int i = 0; i < 8; ++i) d[i] = exilu_f(d[i]);

        // D layout -> B layout for next layer (SWAPX16 mixes).
        v2f q0, q1;
        q0.x = bsel(lo, d[0], d[2]); q0.y = bsel(lo, d[1], d[3]);
        q1.x = bsel(lo, d[4], d[6]); q1.y = bsel(lo, d[5], d[7]);

        // ---- Layer 2: D2 = W2 * h1^T + b2 ----
        v8f e = __builtin_amdgcn_wmma_f32_16x16x4_f32(
            false, a2c0, false, q0, (short)0, c2, false, false);
        e = __builtin_amdgcn_wmma_f32_16x16x4_f32(
            false, a2c1, false, q1, (short)0, e, false, false);
#pragma unroll
        for (int i = 0; i < 8; ++i) e[i] = exilu_f(e[i]);

        q0.x = bsel(lo, e[0], e[2]); q0.y = bsel(lo, e[1], e[3]);
        q1.x = bsel(lo, e[4], e[6]); q1.y = bsel(lo, e[5], e[7]);

        // ---- Layer 3: D3 = W3 * h2^T + b3 ----
        v8f f = __builtin_amdgcn_wmma_f32_16x16x4_f32(
            false, a3c0, false, q0, (short)0, c3, false, false);
        f = __builtin_amdgcn_wmma_f32_16x16x4_f32(
            false, a3c1, false, q1, (short)0, f, false, false);

        // D3(M=feat 0..5, N=row): lane n (0..15) holds out[row n][0..5] in f[0..5].
        if (lo && r0 < rows) {
            float* o = out + r0 * 6;   // 24B stride -> 8B aligned
            v2f s0; s0.x = f[0]; s0.y = f[1];
            v2f s1; s1.x = f[2]; s1.y = f[3];
            v2f s2; s2.x = f[4]; s2.y = f[5];
            __builtin_nontemporal_store(s0, (v2f*)(o + 0));
            __builtin_nontemporal_store(s1, (v2f*)(o + 2));
            __builtin_nontemporal_store(s2, (v2f*)(o + 4));
        }
    }
}

extern "C" void kernel_launch(void* const* d_in, const int* in_sizes, int n_in,
                              void* d_out, int out_size, void* d_ws, size_t ws_size,
                              hipStream_t stream) {
    (void)n_in; (void)out_size; (void)d_ws; (void)ws_size;
    const float* x  = (const float*)d_in[0];
    const float* W1 = (const float*)d_in[1];
    const float* b1 = (const float*)d_in[2];
    const float* W2 = (const float*)d_in[3];
    const float* b2 = (const float*)d_in[4];
    const float* W3 = (const float*)d_in[5];
    const float* b3 = (const float*)d_in[6];
    float* out = (float*)d_out;

    const long rows      = (long)in_sizes[0] / 6;     // B = 8,388,608
    const int  num_tiles = (int)((rows + 15) / 16);   // 524,288 tiles of 16 rows

    const int threads = 256;                          // 8 waves per block
    const int wpb     = threads / 32;
    long want = ((long)num_tiles + wpb - 1) / wpb;
    int blocks = (int)(want < 8192 ? want : 8192);
    if (blocks < 1) blocks = 1;

    LeBRS_mlp_wmma<<<blocks, threads, 0, stream>>>(
        x, W1, b1, W2, b2, W3, b3, out, num_tiles, rows);
}